// LSTMPose_64845416235792
// MI455X (gfx1250) — compile-verified
//
#include <hip/hip_runtime.h>
#include <hip/hip_bf16.h>
#include <cstddef>

#define HWD   36
#define NPIX  (HWD*HWD)     // 1296
#define CIN   64
#define COUT  8
#define PATCH 9
#define NPP   (PATCH*PATCH) // 81
#define OUTD  24
#define FLATD 162           // 2*9*9

typedef float v2f __attribute__((ext_vector_type(2)));
typedef float v8f __attribute__((ext_vector_type(8)));

__device__ __forceinline__ float fast_sigmoid(float x) {
    return 1.0f / (1.0f + __expf(-x));
}
__device__ __forceinline__ float fast_tanh(float x) {
    // tanh(x) = 1 - 2/(exp(2x)+1)
    return 1.0f - 2.0f / (__expf(2.0f * x) + 1.0f);
}

// ---------------------------------------------------------------------------
// Prep: fuse the two 1x1 convs:  Wf[8][64] = w2[8][32] @ w1[32][64],
//                                bf[8]     = w2 @ b1 + b2
// ---------------------------------------------------------------------------
__global__ void lstmpose_prep(const float* __restrict__ w1, const float* __restrict__ b1,
                              const float* __restrict__ w2, const float* __restrict__ b2,
                              float* __restrict__ Wf, float* __restrict__ bf) {
    int t = threadIdx.x;                // 512 threads
    if (t < COUT * CIN) {
        int o = t >> 6, i = t & 63;
        float s = 0.0f;
        #pragma unroll
        for (int m = 0; m < 32; ++m) s += w2[o * 32 + m] * w1[m * 64 + i];
        Wf[o * CIN + i] = s;
    }
    if (t < COUT) {
        float s = b2[t];
        #pragma unroll
        for (int m = 0; m < 32; ++m) s += w2[t * 32 + m] * b1[m];
        bf[t] = s;
    }
}

// ---------------------------------------------------------------------------
// Main fused kernel: one workgroup per sample.
//   Phase 1: encoder GEMM (8x64 per pixel) with V_WMMA_F32_16X16X4_F32 -> LDS
//   Phase 2: 2-layer 1x1 ConvLSTM scan over 16 patches, state in registers
//   Phase 3: 24x162 output matvec
// ---------------------------------------------------------------------------
__global__ void __launch_bounds__(256)
lstmpose_main(const float* __restrict__ color, const float* __restrict__ geo,
              const float* __restrict__ Wf,    const float* __restrict__ bf,
              const float* __restrict__ lw1,   const float* __restrict__ lb1,
              const float* __restrict__ lw2,   const float* __restrict__ lb2,
              const float* __restrict__ ow,    const float* __restrict__ ob,
              float* __restrict__ out) {
    __shared__ float s_enc[COUT][NPIX];   // 41472 B, encoder output
    __shared__ float s_flat[FLATD];       // last h2, flattened [2,9,9]
    __shared__ float s_w1[80], s_b1[8], s_w2[32], s_b2[8];

    const int n     = blockIdx.x;
    const int tid   = threadIdx.x;
    const int lane  = tid & 31;
    const int wave  = tid >> 5;          // 0..7
    const int m     = lane & 15;         // WMMA row (out channel) / pixel-in-tile
    const int khalf = lane >> 4;         // which K pair of the chunk this lane holds

    // Stage LSTM weights into LDS (uniform broadcast reads later).
    if (tid < 80) s_w1[tid] = lw1[tid];
    if (tid < 32) s_w2[tid] = lw2[tid];
    if (tid < 8)  { s_b1[tid] = lb1[tid]; s_b2[tid] = lb2[tid]; }

    // ---- Phase 1: encoder via f32 WMMA -----------------------------------
    // A (16x4 f32 per chunk): lane<16 holds A[m][4kc+0..1], lane>=16 A[m][4kc+2..3].
    // Rows 8..15 are zero padding (COUT=8).
    v2f A[16];
    #pragma unroll
    for (int kc = 0; kc < 16; ++kc) {
        int k0 = 4 * kc + 2 * khalf;
        A[kc].x = (m < COUT) ? Wf[m * CIN + k0]     : 0.0f;
        A[kc].y = (m < COUT) ? Wf[m * CIN + k0 + 1] : 0.0f;
    }
    float bfr[COUT];
    #pragma unroll
    for (int c = 0; c < COUT; ++c) bfr[c] = bf[c];   // uniform -> s_load

    const float* xc = color + (size_t)n * 32 * NPIX;
    const float* xg = geo   + (size_t)n * 32 * NPIX;

    // 81 tiles of 16 pixels; wave-uniform trip counts keep EXEC all-ones at WMMA.
    for (int tile = wave; tile < NPIX / 16; tile += 8) {
        const int p = tile * 16 + m;     // pixel this lane owns within the tile
        v8f acc = {};
        #pragma unroll
        for (int kc = 0; kc < 16; ++kc) {
            const int k0 = 4 * kc + 2 * khalf;   // even, so k0 & k0+1 same tensor
            const float* src = (k0 < 32) ? (xc + (size_t)k0 * NPIX)
                                         : (xg + (size_t)(k0 - 32) * NPIX);
            v2f B;
            B.x = src[p];          // B[K=k0  ][N=p]
            B.y = src[p + NPIX];   // B[K=k0+1][N=p]
            acc = __builtin_amdgcn_wmma_f32_16x16x4_f32(
                      false, A[kc], false, B, (short)0, acc, false, false);
        }
        // D layout: lane<16, VGPR c -> element (M=c, N=p). Only M<8 is real.
        if (khalf == 0) {
            #pragma unroll
            for (int c = 0; c < COUT; ++c)
                s_enc[c][p] = acc[c] + bfr[c];
        }
    }
    __syncthreads();

    // ---- Phase 2: two stacked 1x1 ConvLSTMs over T=16 patches ------------
    if (tid < NPP) {
        const int pi = tid / PATCH, pj = tid % PATCH;
        float h1[2] = {0, 0}, c1[2] = {0, 0}, h2[2] = {0, 0}, c2[2] = {0, 0};
        for (int t = 0; t < 16; ++t) {
            const int bh = t >> 2, bw = t & 3;   // (h_block, w_block) order
            const int p = (bh * PATCH + pi) * HWD + (bw * PATCH + pj);
            float x[COUT];
            #pragma unroll
            for (int k = 0; k < COUT; ++k) x[k] = s_enc[k][p];

            // Layer 1: gates = W1 @ [x; h1] + b1   (gate order: i,f,o,g)
            float g1[8];
            #pragma unroll
            for (int g = 0; g < 8; ++g) {
                float s = s_b1[g];
                #pragma unroll
                for (int k = 0; k < COUT; ++k) s += s_w1[g * 10 + k] * x[k];
                s += s_w1[g * 10 + 8] * h1[0] + s_w1[g * 10 + 9] * h1[1];
                g1[g] = s;
            }
            #pragma unroll
            for (int d = 0; d < 2; ++d) {
                float ig = fast_sigmoid(g1[0 + d]);
                float fg = fast_sigmoid(g1[2 + d]);
                float og = fast_sigmoid(g1[4 + d]);
                float gg = fast_tanh(g1[6 + d]);
                c1[d] = fg * c1[d] + ig * gg;
                h1[d] = og * fast_tanh(c1[d]);
            }

            // Layer 2: gates = W2 @ [h1; h2] + b2
            float g2[8];
            #pragma unroll
            for (int g = 0; g < 8; ++g) {
                float s = s_b2[g];
                s += s_w2[g * 4 + 0] * h1[0] + s_w2[g * 4 + 1] * h1[1];
                s += s_w2[g * 4 + 2] * h2[0] + s_w2[g * 4 + 3] * h2[1];
                g2[g] = s;
            }
            #pragma unroll
            for (int d = 0; d < 2; ++d) {
                float ig = fast_sigmoid(g2[0 + d]);
                float fg = fast_sigmoid(g2[2 + d]);
                float og = fast_sigmoid(g2[4 + d]);
                float gg = fast_tanh(g2[6 + d]);
                c2[d] = fg * c2[d] + ig * gg;
                h2[d] = og * fast_tanh(c2[d]);
            }
        }
        // flatten [2,9,9] -> c*81 + i*9 + j
        s_flat[0 * NPP + tid] = h2[0];
        s_flat[1 * NPP + tid] = h2[1];
    }
    __syncthreads();

    // ---- Phase 3: [162] @ out_w.T + out_b -> [24] ------------------------
    if (tid < OUTD) {
        float s = ob[tid];
        const float* wrow = ow + tid * FLATD;
        #pragma unroll 6
        for (int k = 0; k < FLATD; ++k) s += wrow[k] * s_flat[k];
        out[(size_t)n * OUTD + tid] = s;
    }
}

extern "C" void kernel_launch(void* const* d_in, const int* in_sizes, int n_in,
                              void* d_out, int out_size, void* d_ws, size_t ws_size,
                              hipStream_t stream) {
    const float* color = (const float*)d_in[0];
    const float* geo   = (const float*)d_in[1];
    const float* ew1   = (const float*)d_in[2];
    const float* eb1   = (const float*)d_in[3];
    const float* ew2   = (const float*)d_in[4];
    const float* eb2   = (const float*)d_in[5];
    const float* lw1   = (const float*)d_in[6];
    const float* lb1   = (const float*)d_in[7];
    const float* lw2   = (const float*)d_in[8];
    const float* lb2   = (const float*)d_in[9];
    const float* ow    = (const float*)d_in[10];
    const float* ob    = (const float*)d_in[11];
    float* out = (float*)d_out;

    const int N = in_sizes[0] / (32 * NPIX);   // 2048

    float* Wf = (float*)d_ws;            // 8*64 floats
    float* bf = Wf + COUT * CIN;         // 8 floats

    lstmpose_prep<<<1, 512, 0, stream>>>(ew1, eb1, ew2, eb2, Wf, bf);
    lstmpose_main<<<N, 256, 0, stream>>>(color, geo, Wf, bf,
                                         lw1, lb1, lw2, lb2, ow, ob, out);
}